// ACmix_46608985096484
// MI455X (gfx1250) — compile-verified
//
#include <hip/hip_runtime.h>
#include <hip/hip_bf16.h>
#include <math.h>

typedef float v2f __attribute__((ext_vector_type(2)));
typedef float v8f __attribute__((ext_vector_type(8)));

// ---- problem constants (fixed by setup_inputs) ----
#define BATCH   2
#define CCH     64          // channels
#define DD      8
#define WW      32
#define HH      32
#define SPAT    8192        // d*w*h = H2*W2
#define H2      64
#define W2      128
#define NHEAD   4
#define HDIM    16
#define NCOL    (BATCH*SPAT)        // 16384 gemm columns
#define QKVCOL  (BATCH*HDIM*SPAT)   // 262144 columns for wfc gemm

// ---- workspace offsets (in floats) ----
#define Q_OFF    ((size_t)0)
#define K_OFF    ((size_t)1048576)
#define V_OFF    ((size_t)2097152)
#define PE_OFF   ((size_t)3145728)
#define ATT_OFF  ((size_t)3276800)
#define FALL_OFF ((size_t)4325376)   // f_all: 9 rows x 262144 cols

// =====================================================================
// Kernel 1: q/k/v projections.  D(16x16) = W(16x4) x X(4x16), K=64.
// 12288 tiles total: mtile 0..11 (3 matrices x 4 row-tiles) x 1024 col-tiles.
// One wave per tile, 8 waves per 256-thread block -> 1536 blocks exactly.
// =====================================================================
__global__ void __launch_bounds__(256) qkv_gemm_wmma(
    const float* __restrict__ x,
    const float* __restrict__ w1, const float* __restrict__ b1,
    const float* __restrict__ w2, const float* __restrict__ b2,
    const float* __restrict__ w3, const float* __restrict__ b3,
    float* __restrict__ ws) {
  const int wave = blockIdx.x * 8 + (threadIdx.x >> 5);
  const int lane = threadIdx.x & 31;
  const int half = lane >> 4;           // K half select
  const int ln   = lane & 15;           // M (for A) / N (for B,D)
  const int mtile = wave >> 10;         // 0..11
  const int ntile = wave & 1023;        // 0..1023
  const int which = mtile >> 2;         // 0=q 1=k 2=v
  const int m0    = (mtile & 3) << 4;   // output-channel tile base
  const int col0  = ntile << 4;         // global column base
  const int bat   = col0 >> 13;         // column / 8192
  const int s0    = col0 & 8191;

  const float* W  = (which == 0) ? w1 : (which == 1) ? w2 : w3;
  const float* BI = (which == 0) ? b1 : (which == 1) ? b2 : b3;
  const float* xb = x + ((size_t)bat * CCH) * SPAT;

  v8f acc = {};
#pragma unroll
  for (int kk = 0; kk < 64; kk += 4) {
    const int ka = kk + 2 * half;       // this lane-half's K pair
    v2f a, bv;
    a.x  = W[(m0 + ln) * 64 + ka];      // A[M=ln, K=ka]   (vgpr0)
    a.y  = W[(m0 + ln) * 64 + ka + 1];  // A[M=ln, K=ka+1] (vgpr1)
    bv.x = xb[(size_t)ka * SPAT + s0 + ln];        // B[K=ka, N=ln]
    bv.y = xb[(size_t)(ka + 1) * SPAT + s0 + ln];  // B[K=ka+1, N=ln]
    acc = __builtin_amdgcn_wmma_f32_16x16x4_f32(
        false, a, false, bv, (short)0, acc, false, false);
  }

  float* dst = ws + ((which == 0) ? Q_OFF : (which == 1) ? K_OFF : V_OFF)
                  + ((size_t)bat * CCH) * SPAT;
#pragma unroll
  for (int r = 0; r < 8; ++r) {
    const int M = m0 + r + 8 * half;    // D layout: vgpr r -> M=r (+8 hi half)
    dst[(size_t)M * SPAT + s0 + ln] = acc[r] + BI[M];
  }
}

// =====================================================================
// Kernel 2: position encoding pe[c][s], c<16, s<8192
// =====================================================================
__global__ void __launch_bounds__(256) pe_kernel(
    const float* __restrict__ wp, const float* __restrict__ bp,
    float* __restrict__ ws) {
  const int idx = blockIdx.x * blockDim.x + threadIdx.x;  // 0..131071
  const int c = idx >> 13;
  const int s = idx & 8191;
  const int di = s >> 10;
  const int rem = s & 1023;
  const int wi = rem >> 5;
  const int hi = rem & 31;
  const float ld = -1.f + 2.f * (float)di / 7.f;
  const float lw = -1.f + 2.f * (float)wi / 31.f;
  const float lh = -1.f + 2.f * (float)hi / 31.f;
  ws[PE_OFF + idx] = wp[c * 3 + 0] * ld + wp[c * 3 + 1] * lw
                   + wp[c * 3 + 2] * lh + bp[c];
}

__device__ __forceinline__ int reflect_idx(int m, int n) {
  // ReflectionPad (pad=3 < n): single reflection suffices
  m = (m < 0) ? -m : m;
  m = (m >= n) ? (2 * n - 2 - m) : m;
  return m;
}

// =====================================================================
// Kernel 3: local 7x7 attention with reflect padding + relative PE.
// One thread per (n, p), n = b*4+head in [0,8), p in [0,8192).
// =====================================================================
__global__ void __launch_bounds__(256) attn_kernel(float* __restrict__ ws) {
  const int idx = blockIdx.x * blockDim.x + threadIdx.x;  // 0..65535
  const int n = idx >> 13;
  const int p = idx & 8191;
  const int y = p >> 7;        // H2 row
  const int xcol = p & 127;    // W2 col
  const int bb = n >> 2;
  const int head = n & 3;
  const size_t chbase = ((size_t)bb * CCH + head * HDIM) * SPAT;
  const float* qp = ws + Q_OFF + chbase;
  const float* kp = ws + K_OFF + chbase;
  const float* vp = ws + V_OFF + chbase;
  const float* pep = ws + PE_OFF;

  float qv[HDIM];
  float qdotpe = 0.f;
#pragma unroll
  for (int c = 0; c < HDIM; ++c) {
    qv[c] = qp[(size_t)c * SPAT + p] * 0.25f;       // scaling = 16^-0.5
    qdotpe += qv[c] * pep[(size_t)c * SPAT + p];
  }

  float logit[49];
  float mx = -3.4e38f;
  int t = 0;
  for (int di = -3; di <= 3; ++di) {
    const int yy = reflect_idx(y + di, H2);
    for (int dj = -3; dj <= 3; ++dj) {
      const int xx = reflect_idx(xcol + dj, W2);
      const int pp = yy * W2 + xx;
      float dk = 0.f, dr = 0.f;
#pragma unroll
      for (int c = 0; c < HDIM; ++c) {
        dk += qv[c] * kp[(size_t)c * SPAT + pp];
        dr += qv[c] * pep[(size_t)c * SPAT + pp];
      }
      const float l = dk - dr + qdotpe;
      logit[t++] = l;
      mx = fmaxf(mx, l);
    }
  }
  float sum = 0.f;
#pragma unroll
  for (int k = 0; k < 49; ++k) { logit[k] = __expf(logit[k] - mx); sum += logit[k]; }
  const float inv = 1.f / sum;

  float acc[HDIM];
#pragma unroll
  for (int c = 0; c < HDIM; ++c) acc[c] = 0.f;
  t = 0;
  for (int di = -3; di <= 3; ++di) {
    const int yy = reflect_idx(y + di, H2);
    for (int dj = -3; dj <= 3; ++dj) {
      const int xx = reflect_idx(xcol + dj, W2);
      const int pp = yy * W2 + xx;
      const float wgt = logit[t++] * inv;
#pragma unroll
      for (int c = 0; c < HDIM; ++c) acc[c] += wgt * vp[(size_t)c * SPAT + pp];
    }
  }
#pragma unroll
  for (int c = 0; c < HDIM; ++c)
    ws[ATT_OFF + chbase + (size_t)c * SPAT + p] = acc[c];
}

// =====================================================================
// Kernel 4: f_all[kidx][col] = wfc(9x12) x qkv(12 x 262144), WMMA K=12.
// col = (b*16 + hd)*8192 + s.  16384 tiles, 8 waves/block -> 2048 blocks.
// =====================================================================
__global__ void __launch_bounds__(256) fall_gemm_wmma(
    const float* __restrict__ wfc, const float* __restrict__ ws_in,
    float* __restrict__ ws_out) {
  const int wave = blockIdx.x * 8 + (threadIdx.x >> 5);
  const int lane = threadIdx.x & 31;
  const int half = lane >> 4;
  const int ln   = lane & 15;
  const int col0 = wave << 4;
  const int bat  = col0 >> 17;          // /131072
  const int hd   = (col0 >> 13) & 15;
  const int s0   = col0 & 8191;

  v8f acc = {};
#pragma unroll
  for (int kk = 0; kk < 12; kk += 4) {
    const int ka = kk + 2 * half;       // 0..11
    v2f a, bv;
    a.x = (ln < 9) ? wfc[ln * 12 + ka] : 0.f;
    a.y = (ln < 9) ? wfc[ln * 12 + ka + 1] : 0.f;
    // B row m: qkv[b, m, hd, s] = {q,k,v chunk m/4}[(b*64 + (m%4)*16 + hd)*8192 + s]
    {
      const int m = ka;
      const float* src = ws_in + ((m >> 2) == 0 ? Q_OFF : (m >> 2) == 1 ? K_OFF : V_OFF);
      bv.x = src[((size_t)bat * CCH + (m & 3) * HDIM + hd) * SPAT + s0 + ln];
    }
    {
      const int m = ka + 1;
      const float* src = ws_in + ((m >> 2) == 0 ? Q_OFF : (m >> 2) == 1 ? K_OFF : V_OFF);
      bv.y = src[((size_t)bat * CCH + (m & 3) * HDIM + hd) * SPAT + s0 + ln];
    }
    acc = __builtin_amdgcn_wmma_f32_16x16x4_f32(
        false, a, false, bv, (short)0, acc, false, false);
  }
#pragma unroll
  for (int r = 0; r < 8; ++r) {
    const int row = r + 8 * half;       // kidx 0..15, only 0..8 valid
    if (row < 9)
      ws_out[FALL_OFF + (size_t)row * QKVCOL + col0 + ln] = acc[r];
  }
}

// =====================================================================
// Kernel 5: grouped 3x3 conv over (d, w*h) using wdep exactly + combine.
// out = rate1*out_att + rate2*out_conv.  One thread per output element.
// =====================================================================
__global__ void __launch_bounds__(256) finalize_kernel(
    const float* __restrict__ ws, const float* __restrict__ wdep,
    const float* __restrict__ bdep, const float* __restrict__ rate1,
    const float* __restrict__ rate2, float* __restrict__ out) {
  const int idx = blockIdx.x * blockDim.x + threadIdx.x;  // 0..1048575
  const int bb = idx >> 19;
  const int rest = idx & 524287;
  const int o = rest >> 13;
  const int sp = rest & 8191;
  const int y = sp >> 10;               // d axis (conv "height", 8)
  const int xcol = sp & 1023;           // w*h flat (conv "width", 1024)
  const int g = o >> 2;                 // group (feature_group_count = 16)

  float conv = bdep[o];
#pragma unroll
  for (int i = 0; i < 9; ++i) {
    const int c = g * 9 + i;            // f_conv channel = kidx*16 + hd
    const int kidx = c >> 4;
    const int hd = c & 15;
    const float* fa = ws + FALL_OFF + (size_t)kidx * QKVCOL
                    + ((size_t)bb * HDIM + hd) * SPAT;
    const float* wk = wdep + ((size_t)o * 9 + i) * 9;
#pragma unroll
    for (int ki = 0; ki < 3; ++ki) {
      const int yy = y + ki - 1;
      if (yy < 0 || yy >= DD) continue;
#pragma unroll
      for (int kj = 0; kj < 3; ++kj) {
        const int xx = xcol + kj - 1;
        if (xx < 0 || xx >= 1024) continue;
        conv += wk[ki * 3 + kj] * fa[(size_t)yy * 1024 + xx];
      }
    }
  }
  out[idx] = rate1[0] * ws[ATT_OFF + idx] + rate2[0] * conv;
}

// =====================================================================
extern "C" void kernel_launch(void* const* d_in, const int* in_sizes, int n_in,
                              void* d_out, int out_size, void* d_ws, size_t ws_size,
                              hipStream_t stream) {
  const float* x    = (const float*)d_in[0];
  const float* w1   = (const float*)d_in[1];
  const float* b1   = (const float*)d_in[2];
  const float* w2   = (const float*)d_in[3];
  const float* b2   = (const float*)d_in[4];
  const float* w3   = (const float*)d_in[5];
  const float* b3   = (const float*)d_in[6];
  const float* wp   = (const float*)d_in[7];
  const float* bp   = (const float*)d_in[8];
  const float* wfc  = (const float*)d_in[9];
  const float* wdep = (const float*)d_in[10];
  const float* bdep = (const float*)d_in[11];
  const float* r1   = (const float*)d_in[12];
  const float* r2   = (const float*)d_in[13];
  float* ws  = (float*)d_ws;
  float* out = (float*)d_out;

  // 1) q/k/v GEMMs (WMMA): 12288 tiles, 8 waves/block
  qkv_gemm_wmma<<<1536, 256, 0, stream>>>(x, w1, b1, w2, b2, w3, b3, ws);
  // 2) position encoding: 131072 elems
  pe_kernel<<<512, 256, 0, stream>>>(wp, bp, ws);
  // 3) attention: 65536 threads
  attn_kernel<<<256, 256, 0, stream>>>(ws);
  // 4) wfc GEMM (WMMA): 16384 tiles, 8 waves/block
  fall_gemm_wmma<<<2048, 256, 0, stream>>>(wfc, ws, ws);
  // 5) grouped conv + combine: 1048576 elems
  finalize_kernel<<<4096, 256, 0, stream>>>(ws, wdep, bdep, r1, r2, out);
}